// RecommendationModel_27092653703485
// MI455X (gfx1250) — compile-verified
//
#include <hip/hip_runtime.h>
#include <hip/hip_bf16.h>

typedef float v2f __attribute__((ext_vector_type(2)));
typedef float v8f __attribute__((ext_vector_type(8)));

#define NU_N 100000
#define NM_N 50000
#define E_N  1000000
#define L_N  250000
#define H_N  64
#define F_N  24

// Swizzled weight layout so each lane's WMMA B-operand pair {W[k][n], W[k+1][n]}
// is an aligned contiguous 8B in LDS:
//   dst(k, n) = (k>>2)*256 + ((k>>1)&1)*128 + n*2 + (k&1)
__device__ __forceinline__ int wswz(int k, int n) {
  return ((k >> 2) << 8) + (((k >> 1) & 1) << 7) + (n << 1) + (k & 1);
}

// ---------------------------------------------------------------------------
// zero fill
// ---------------------------------------------------------------------------
__global__ void zero_kernel(float* __restrict__ p, long long n) {
  long long i = (long long)blockIdx.x * blockDim.x + threadIdx.x;
  long long stride = (long long)gridDim.x * blockDim.x;
  for (; i < n; i += stride) p[i] = 0.0f;
}

// ---------------------------------------------------------------------------
// xm0 = movie_x[NM,24] @ W[24,64] + b + movie_emb   (WMMA f32 16x16x4, K=24)
// ---------------------------------------------------------------------------
__global__ __launch_bounds__(256) void movie_init_kernel(
    const float* __restrict__ mx, const float* __restrict__ w,
    const float* __restrict__ b, const float* __restrict__ memb,
    float* __restrict__ xm, int N) {
  __shared__ float sW[F_N * H_N];
  for (int i = threadIdx.x; i < F_N * H_N; i += 256) {
    int k = i >> 6, n = i & 63;
    sW[wswz(k, n)] = w[i];
  }
  __syncthreads();
  const int wave = threadIdx.x >> 5;
  const int lane = threadIdx.x & 31;
  const int row0 = (blockIdx.x * 8 + wave) * 16;
  if (row0 >= N) return;
  const int nl = lane & 15;
  const int half = lane >> 4;
  const int kHi = half * 2;
  const int r = row0 + nl;
  const float* aRow = mx + (size_t)r * F_N;

  v8f c[4];
  {
    float b0 = b[nl], b1 = b[16 + nl], b2 = b[32 + nl], b3 = b[48 + nl];
#pragma unroll
    for (int j = 0; j < 8; ++j) { c[0][j] = b0; c[1][j] = b1; c[2][j] = b2; c[3][j] = b3; }
  }
#pragma unroll
  for (int ks = 0; ks < F_N / 4; ++ks) {
    v2f a = *reinterpret_cast<const v2f*>(aRow + ks * 4 + kHi);
    const float* bp = sW + (ks << 8) + (half << 7) + (nl << 1);
#pragma unroll
    for (int ct = 0; ct < 4; ++ct) {
      v2f bb = *reinterpret_cast<const v2f*>(bp + ct * 32);
      c[ct] = __builtin_amdgcn_wmma_f32_16x16x4_f32(false, a, false, bb,
                                                    (short)0, c[ct], false, false);
    }
  }
  const int rowStore = row0 + half * 8;
#pragma unroll
  for (int j = 0; j < 8; ++j) {
    const float* er = memb + (size_t)(rowStore + j) * H_N + nl;
    float* yr = xm + (size_t)(rowStore + j) * H_N + nl;
    yr[0]  = c[0][j] + er[0];
    yr[16] = c[1][j] + er[16];
    yr[32] = c[2][j] + er[32];
    yr[48] = c[3][j] + er[48];
  }
}

// ---------------------------------------------------------------------------
// scatter: agg[dst[e]] += xsrc[src[e]]; cnt[dst[e]] += 1   (f32 atomics, L2)
// one thread per (edge, 16B chunk)
// ---------------------------------------------------------------------------
__global__ void scatter_kernel(const float* __restrict__ xsrc,
                               const int* __restrict__ src,
                               const int* __restrict__ dst,
                               float* __restrict__ agg,
                               float* __restrict__ cntv, int E) {
  long long t = (long long)blockIdx.x * blockDim.x + threadIdx.x;
  long long total = (long long)E * 16;
  if (t >= total) return;
  int e = (int)(t >> 4);
  int q = (int)(t & 15);
  int s = src[e], d = dst[e];
  const float4 v = *reinterpret_cast<const float4*>(xsrc + (size_t)s * H_N + q * 4);
  float* a = agg + (size_t)d * H_N + q * 4;
  __hip_atomic_fetch_add(a + 0, v.x, __ATOMIC_RELAXED, __HIP_MEMORY_SCOPE_AGENT);
  __hip_atomic_fetch_add(a + 1, v.y, __ATOMIC_RELAXED, __HIP_MEMORY_SCOPE_AGENT);
  __hip_atomic_fetch_add(a + 2, v.z, __ATOMIC_RELAXED, __HIP_MEMORY_SCOPE_AGENT);
  __hip_atomic_fetch_add(a + 3, v.w, __ATOMIC_RELAXED, __HIP_MEMORY_SCOPE_AGENT);
  if (q == 0)
    __hip_atomic_fetch_add(cntv + d, 1.0f, __ATOMIC_RELAXED, __HIP_MEMORY_SCOPE_AGENT);
}

// ---------------------------------------------------------------------------
// y = (agg * 1/max(cnt,1)) @ wl + bl + xdst @ wr    (WMMA f32 16x16x4, K=64)
// ---------------------------------------------------------------------------
__global__ __launch_bounds__(256) void sage_gemm_kernel(
    const float* __restrict__ agg, const float* __restrict__ cnt,
    const float* __restrict__ xdst, const float* __restrict__ wl,
    const float* __restrict__ bl, const float* __restrict__ wr,
    float* __restrict__ y, int N) {
  __shared__ float sW[2 * H_N * H_N];
  for (int i = threadIdx.x; i < H_N * H_N; i += 256) {
    int k = i >> 6, n = i & 63;
    int d = wswz(k, n);
    sW[d] = wl[i];
    sW[H_N * H_N + d] = wr[i];
  }
  __syncthreads();
  const int wave = threadIdx.x >> 5;
  const int lane = threadIdx.x & 31;
  const int row0 = (blockIdx.x * 8 + wave) * 16;
  if (row0 >= N) return;
  const int nl = lane & 15;
  const int half = lane >> 4;
  const int kHi = half * 2;
  const int r = row0 + nl;
  const float s = 1.0f / fmaxf(cnt[r], 1.0f);
  const float* aggRow = agg + (size_t)r * H_N;
  const float* xRow = xdst + (size_t)r * H_N;

  v8f c[4];
  {
    float b0 = bl[nl], b1 = bl[16 + nl], b2 = bl[32 + nl], b3 = bl[48 + nl];
#pragma unroll
    for (int j = 0; j < 8; ++j) { c[0][j] = b0; c[1][j] = b1; c[2][j] = b2; c[3][j] = b3; }
  }
#pragma unroll
  for (int ks = 0; ks < H_N / 4; ++ks) {
    v2f am = *reinterpret_cast<const v2f*>(aggRow + ks * 4 + kHi);
    v2f ax = *reinterpret_cast<const v2f*>(xRow + ks * 4 + kHi);
    am = am * s;
    const float* bL = sW + (ks << 8) + (half << 7) + (nl << 1);
    const float* bR = bL + H_N * H_N;
#pragma unroll
    for (int ct = 0; ct < 4; ++ct) {
      v2f bwl = *reinterpret_cast<const v2f*>(bL + ct * 32);
      v2f bwr = *reinterpret_cast<const v2f*>(bR + ct * 32);
      c[ct] = __builtin_amdgcn_wmma_f32_16x16x4_f32(false, am, false, bwl,
                                                    (short)0, c[ct], false, false);
      c[ct] = __builtin_amdgcn_wmma_f32_16x16x4_f32(false, ax, false, bwr,
                                                    (short)0, c[ct], false, false);
    }
  }
  const int rowStore = row0 + half * 8;
#pragma unroll
  for (int j = 0; j < 8; ++j) {
    float* yr = y + (size_t)(rowStore + j) * H_N + nl;
    yr[0] = c[0][j]; yr[16] = c[1][j]; yr[32] = c[2][j]; yr[48] = c[3][j];
  }
}

// ---------------------------------------------------------------------------
// BatchNorm: column sums / sums of squares -> stats[0:64]=sum, [64:128]=sumsq
// ---------------------------------------------------------------------------
__global__ __launch_bounds__(256) void bn_stats_kernel(const float* __restrict__ y,
                                                       long long n,
                                                       float* __restrict__ stats) {
  __shared__ float ls[256], ls2[256];
  float s = 0.0f, s2 = 0.0f;
  long long i = (long long)blockIdx.x * 256 + threadIdx.x;
  long long stride = (long long)gridDim.x * 256;
  for (; i < n; i += stride) { float v = y[i]; s += v; s2 += v * v; }
  ls[threadIdx.x] = s; ls2[threadIdx.x] = s2;
  __syncthreads();
  if (threadIdx.x < 64) {
    s  = ls[threadIdx.x]  + ls[threadIdx.x + 64]  + ls[threadIdx.x + 128]  + ls[threadIdx.x + 192];
    s2 = ls2[threadIdx.x] + ls2[threadIdx.x + 64] + ls2[threadIdx.x + 128] + ls2[threadIdx.x + 192];
    __hip_atomic_fetch_add(&stats[threadIdx.x],      s,  __ATOMIC_RELAXED, __HIP_MEMORY_SCOPE_AGENT);
    __hip_atomic_fetch_add(&stats[64 + threadIdx.x], s2, __ATOMIC_RELAXED, __HIP_MEMORY_SCOPE_AGENT);
  }
}

__global__ void bn_finalize_kernel(float* __restrict__ stats, float invN) {
  int t = threadIdx.x;  // 64 threads
  float mu = stats[t] * invN;
  float var = stats[64 + t] * invN - mu * mu;
  stats[t] = mu;
  stats[64 + t] = rsqrtf(var + 1e-5f);
}

__global__ void bn_apply_kernel(const float* __restrict__ y, long long n,
                                const float* __restrict__ stats,
                                const float* __restrict__ g,
                                const float* __restrict__ b,
                                float* __restrict__ xout) {
  long long i = (long long)blockIdx.x * blockDim.x + threadIdx.x;
  long long stride = (long long)gridDim.x * blockDim.x;
  for (; i < n; i += stride) {
    int col = (int)(i & 63);
    float o = g[col] * (y[i] - stats[col]) * stats[64 + col] + b[col];
    xout[i] = fmaxf(o, 0.0f);
  }
}

// ---------------------------------------------------------------------------
// classifier: out[e] = relu(concat(xu[src[e]], xm[dst[e]]) @ W1 + b1) @ W2 + b2
// WMMA K=128, then ReLU * w2 dot-reduce with 16-lane shuffles.
// ---------------------------------------------------------------------------
__global__ __launch_bounds__(256) void cls_kernel(
    const float* __restrict__ xu, const float* __restrict__ xm,
    const int* __restrict__ lsrc, const int* __restrict__ ldst,
    const float* __restrict__ w1, const float* __restrict__ b1,
    const float* __restrict__ w2, const float* __restrict__ b2,
    float* __restrict__ out, int L) {
  __shared__ float sW1[2 * H_N * H_N];
  __shared__ float sW2[H_N];
  for (int i = threadIdx.x; i < 2 * H_N * H_N; i += 256) {
    int k = i >> 6, n = i & 63;
    sW1[wswz(k, n)] = w1[i];
  }
  if (threadIdx.x < H_N) sW2[threadIdx.x] = w2[threadIdx.x];
  __syncthreads();
  const int wave = threadIdx.x >> 5;
  const int lane = threadIdx.x & 31;
  const int row0 = (blockIdx.x * 8 + wave) * 16;
  if (row0 >= L) return;
  const int nl = lane & 15;
  const int half = lane >> 4;
  const int kHi = half * 2;
  const int e = row0 + nl;
  const float* pu = xu + (size_t)lsrc[e] * H_N;
  const float* pm = xm + (size_t)ldst[e] * H_N;

  v8f c[4];
  {
    float b0 = b1[nl], bb1 = b1[16 + nl], b2v = b1[32 + nl], b3 = b1[48 + nl];
#pragma unroll
    for (int j = 0; j < 8; ++j) { c[0][j] = b0; c[1][j] = bb1; c[2][j] = b2v; c[3][j] = b3; }
  }
#pragma unroll
  for (int ks = 0; ks < H_N / 4; ++ks) {
    v2f a = *reinterpret_cast<const v2f*>(pu + ks * 4 + kHi);
    const float* bp = sW1 + (ks << 8) + (half << 7) + (nl << 1);
#pragma unroll
    for (int ct = 0; ct < 4; ++ct) {
      v2f bb = *reinterpret_cast<const v2f*>(bp + ct * 32);
      c[ct] = __builtin_amdgcn_wmma_f32_16x16x4_f32(false, a, false, bb,
                                                    (short)0, c[ct], false, false);
    }
  }
#pragma unroll
  for (int ks = 0; ks < H_N / 4; ++ks) {
    v2f a = *reinterpret_cast<const v2f*>(pm + ks * 4 + kHi);
    const float* bp = sW1 + ((ks + 16) << 8) + (half << 7) + (nl << 1);
#pragma unroll
    for (int ct = 0; ct < 4; ++ct) {
      v2f bb = *reinterpret_cast<const v2f*>(bp + ct * 32);
      c[ct] = __builtin_amdgcn_wmma_f32_16x16x4_f32(false, a, false, bb,
                                                    (short)0, c[ct], false, false);
    }
  }
  // relu(h) @ w2
  v8f acc;
#pragma unroll
  for (int j = 0; j < 8; ++j) acc[j] = 0.0f;
#pragma unroll
  for (int ct = 0; ct < 4; ++ct) {
    float wv = sW2[ct * 16 + nl];
#pragma unroll
    for (int j = 0; j < 8; ++j) acc[j] += fmaxf(c[ct][j], 0.0f) * wv;
  }
#pragma unroll
  for (int off = 1; off < 16; off <<= 1) {
#pragma unroll
    for (int j = 0; j < 8; ++j) acc[j] += __shfl_xor(acc[j], off);
  }
  if (nl == 0) {
    float b2v = b2[0];
    int base = row0 + half * 8;
#pragma unroll
    for (int j = 0; j < 8; ++j) out[base + j] = acc[j] + b2v;
  }
}

// ---------------------------------------------------------------------------
// host launcher
// ---------------------------------------------------------------------------
extern "C" void kernel_launch(void* const* d_in, const int* in_sizes, int n_in,
                              void* d_out, int out_size, void* d_ws, size_t ws_size,
                              hipStream_t stream) {
  (void)in_sizes; (void)n_in; (void)out_size; (void)ws_size;

  const float* movie_x     = (const float*)d_in[0];
  const float* user_emb    = (const float*)d_in[1];
  const float* movie_emb   = (const float*)d_in[2];
  const float* movie_lin_w = (const float*)d_in[3];
  const float* movie_lin_b = (const float*)d_in[4];
  // per layer l: rates (wl,bl,wr), rev (wl,bl,wr), bn user(g,b), bn movie(g,b)
  const float* W[2][2][3];   // [layer][0=rates,1=rev][wl,bl,wr]
  const float* BN[2][2][2];  // [layer][0=user,1=movie][g,b]
  int idx = 5;
  for (int l = 0; l < 2; ++l) {
    for (int et = 0; et < 2; ++et)
      for (int k = 0; k < 3; ++k) W[l][et][k] = (const float*)d_in[idx++];
    for (int nt = 0; nt < 2; ++nt)
      for (int k = 0; k < 2; ++k) BN[l][nt][k] = (const float*)d_in[idx++];
  }
  const float* cls_w1 = (const float*)d_in[25];
  const float* cls_b1 = (const float*)d_in[26];
  const float* cls_w2 = (const float*)d_in[27];
  const float* cls_b2 = (const float*)d_in[28];
  const int* rates_src = (const int*)d_in[31];
  const int* rates_dst = (const int*)d_in[32];
  const int* label_src = (const int*)d_in[33];
  const int* label_dst = (const int*)d_in[34];
  float* out = (float*)d_out;

  // workspace layout (floats)
  float* ws = (float*)d_ws;
  const size_t NUH = (size_t)NU_N * H_N;
  const size_t NMH = (size_t)NM_N * H_N;
  float* xm0   = ws;
  float* xm1   = xm0 + NMH;
  float* xm2   = xm1 + NMH;
  float* xu1   = xm2 + NMH;
  float* xu2   = xu1 + NUH;
  float* agg   = xu2 + NUH;
  float* ybuf  = agg + NUH;
  float* cntb  = ybuf + NUH;
  float* stats = cntb + NU_N;

  const int TB = 256;

  // xm0 = movie_x @ W + b + movie_emb
  movie_init_kernel<<<(NM_N / 16 + 7) / 8, TB, 0, stream>>>(
      movie_x, movie_lin_w, movie_lin_b, movie_emb, xm0, NM_N);

  const float* xu_cur = user_emb;
  const float* xm_cur = xm0;
  float* xu_next[2] = {xu1, xu2};
  float* xm_next[2] = {xm1, xm2};

  for (int l = 0; l < 2; ++l) {
    // ---- user update: agg over rev edges (movie -> user)
    zero_kernel<<<1024, TB, 0, stream>>>(agg, (long long)NUH);
    zero_kernel<<<128, TB, 0, stream>>>(cntb, (long long)NU_N);
    zero_kernel<<<1, 128, 0, stream>>>(stats, 128);
    scatter_kernel<<<(int)(((long long)E_N * 16 + TB - 1) / TB), TB, 0, stream>>>(
        xm_cur, rates_dst, rates_src, agg, cntb, E_N);
    sage_gemm_kernel<<<(NU_N / 16 + 7) / 8, TB, 0, stream>>>(
        agg, cntb, xu_cur, W[l][1][0], W[l][1][1], W[l][1][2], ybuf, NU_N);
    bn_stats_kernel<<<512, TB, 0, stream>>>(ybuf, (long long)NUH, stats);
    bn_finalize_kernel<<<1, 64, 0, stream>>>(stats, 1.0f / (float)NU_N);
    bn_apply_kernel<<<1024, TB, 0, stream>>>(ybuf, (long long)NUH, stats,
                                             BN[l][0][0], BN[l][0][1], xu_next[l]);

    // ---- movie update: agg over rates edges (user -> movie), uses OLD xu
    zero_kernel<<<1024, TB, 0, stream>>>(agg, (long long)NMH);
    zero_kernel<<<128, TB, 0, stream>>>(cntb, (long long)NM_N);
    zero_kernel<<<1, 128, 0, stream>>>(stats, 128);
    scatter_kernel<<<(int)(((long long)E_N * 16 + TB - 1) / TB), TB, 0, stream>>>(
        xu_cur, rates_src, rates_dst, agg, cntb, E_N);
    sage_gemm_kernel<<<(NM_N / 16 + 7) / 8, TB, 0, stream>>>(
        agg, cntb, xm_cur, W[l][0][0], W[l][0][1], W[l][0][2], ybuf, NM_N);
    bn_stats_kernel<<<512, TB, 0, stream>>>(ybuf, (long long)NMH, stats);
    bn_finalize_kernel<<<1, 64, 0, stream>>>(stats, 1.0f / (float)NM_N);
    bn_apply_kernel<<<1024, TB, 0, stream>>>(ybuf, (long long)NMH, stats,
                                             BN[l][1][0], BN[l][1][1], xm_next[l]);

    xu_cur = xu_next[l];
    xm_cur = xm_next[l];
  }

  // classifier
  cls_kernel<<<(L_N / 16 + 7) / 8, TB, 0, stream>>>(
      xu_cur, xm_cur, label_src, label_dst, cls_w1, cls_b1, cls_w2, cls_b2, out, L_N);
}